// Lfm2MoeSparseMoeBlock_18734647345715
// MI455X (gfx1250) — compile-verified
//
#include <hip/hip_runtime.h>

typedef float v2f __attribute__((ext_vector_type(2)));
typedef float v8f __attribute__((ext_vector_type(8)));

constexpr int E    = 16;
constexpr int H    = 1024;
constexpr int I    = 512;
constexpr int T    = 1024;   // B*S
constexpr int TOPK = 4;

constexpr int MT   = 32;     // M-tile rows per block (2x16 sub-tiles)
constexpr int KC   = 512;    // K-chunk staged in LDS for phase A
constexpr int LSTR = 516;    // padded LDS stride (stride%64 words == 4 -> bank spread)

// ---------------- Router: 1 wave per token ----------------
__global__ __launch_bounds__(32) void router_kernel(
    const float* __restrict__ x, const float* __restrict__ gw,
    const float* __restrict__ ebias,
    int* __restrict__ sel, float* __restrict__ rw)
{
    const int t    = blockIdx.x;
    const int lane = threadIdx.x;
    const int e    = lane & 15;
    const int half = lane >> 4;

    const float* xr = x  + (size_t)t * H + half * (H / 2);
    const float* wr = gw + (size_t)e * H + half * (H / 2);
    float acc = 0.f;
    #pragma unroll 4
    for (int i = 0; i < H / 2; ++i) acc += xr[i] * wr[i];
    acc += __shfl_xor(acc, 16, 32);          // combine halves

    __shared__ float score[16];
    __shared__ float biased[16];
    if (lane < 16) {
        float s = 1.f / (1.f + __expf(-acc));
        score[e]  = s;
        biased[e] = s + ebias[e];
    }
    __syncthreads();

    if (lane == 0) {
        int chosen[TOPK];
        float wsum = 0.f;
        unsigned mask = 0;
        for (int k = 0; k < TOPK; ++k) {
            int best = 0; float bv = -1e30f;
            for (int j = 0; j < E; ++j)
                if (!((mask >> j) & 1u) && biased[j] > bv) { bv = biased[j]; best = j; }
            mask |= 1u << best;
            chosen[k] = best;
            wsum += score[best];
        }
        float inv = 1.f / (wsum + 1e-6f);
        for (int k = 0; k < TOPK; ++k) {
            sel[t * TOPK + k] = chosen[k];
            rw [t * TOPK + k] = score[chosen[k]] * inv;   // ROUTED_SCALING_FACTOR = 1
        }
    }
}

// ---------------- Zero per-expert counts ----------------
__global__ __launch_bounds__(32) void zero_kernel(int* __restrict__ counts)
{
    if (threadIdx.x < E) counts[threadIdx.x] = 0;
}

// ---------------- Build per-expert token lists ----------------
__global__ __launch_bounds__(256) void scatter_kernel(
    const int* __restrict__ sel, int* __restrict__ counts,
    int* __restrict__ ltok, int* __restrict__ lslot)
{
    int t = blockIdx.x * 256 + threadIdx.x;
    if (t >= T) return;
    for (int k = 0; k < TOPK; ++k) {
        int e   = sel[t * TOPK + k];
        int pos = atomicAdd(&counts[e], 1);   // position only; results row-independent
        ltok [e * T + pos] = t;
        lslot[e * T + pos] = k;
    }
}

// ---------------- Fused expert MLP: silu(xWg)*xWu @ Wd ----------------
// grid.x = E * 32 (max 32-row m-tiles); block = 256 (8 waves).
// LDS: one 32x516 float buffer, used as X-stage in phase A, then re-used as h-tile.
__global__ __launch_bounds__(256) void expert_kernel(
    const float* __restrict__ x,
    const float* __restrict__ w_gate,
    const float* __restrict__ w_up,
    const float* __restrict__ w_down,
    const float* __restrict__ rw,
    const int*  __restrict__ counts,
    const int*  __restrict__ ltok,
    const int*  __restrict__ lslot,
    float* __restrict__ ybuf)
{
    extern __shared__ float buf[];       // MT * LSTR floats (~66 KB): xs, then hb
    __shared__ int   trow[MT];
    __shared__ int   srow[MT];
    __shared__ float wrow[MT];

    const int e     = blockIdx.x >> 5;
    const int mt    = blockIdx.x & 31;
    const int cnt   = counts[e];
    const int mbase = mt * MT;
    if (mbase >= cnt) return;                 // uniform exit: EXEC stays full
    const int mrows = min(MT, cnt - mbase);

    const int tid = threadIdx.x;
    if (tid < MT) {
        int m   = tid;
        int src = e * T + mbase + (m < mrows ? m : 0);   // pad with row 0 (valid data)
        int tok = ltok[src];
        int slt = lslot[src];
        trow[m] = tok;
        srow[m] = slt;
        wrow[m] = (m < mrows) ? rw[tok * TOPK + slt] : 0.f;
    }
    __syncthreads();

    const int wave  = tid >> 5;
    const int lane  = tid & 31;
    const int lhalf = lane >> 4;   // K-half of the A/B fragments
    const int lidx  = lane & 15;   // M (A) / N (B,C,D) index

    // ---- Phase A: wave owns 64 I-columns; g & u for both 16-row sub-tiles ----
    const int c0 = wave * 64;
    const float* Wg = w_gate + (size_t)e * H * I;
    const float* Wu = w_up   + (size_t)e * H * I;

    v8f cg[2][4], cu[2][4];                 // [m-subtile][n-tile]
    #pragma unroll
    for (int ms = 0; ms < 2; ++ms)
        #pragma unroll
        for (int nt = 0; nt < 4; ++nt) { cg[ms][nt] = (v8f){}; cu[ms][nt] = (v8f){}; }

    for (int kc = 0; kc < H; kc += KC) {
        // stage gathered X chunk: MT rows x KC cols, float4 vectorized
        for (int i = tid; i < MT * (KC / 4); i += 256) {
            int m = i >> 7, c4 = (i & (KC / 4 - 1)) * 4;
            const float4 v = *(const float4*)(x + (size_t)trow[m] * H + kc + c4);
            *(float4*)(buf + m * LSTR + c4) = v;
        }
        __syncthreads();

        #pragma unroll
        for (int nt = 0; nt < 4; ++nt) {
            const int n0 = c0 + nt * 16;
            const float* a0p = buf + lidx * LSTR + 2 * lhalf;
            const float* a1p = a0p + 16 * LSTR;
            const float* wgp = Wg + (size_t)(kc + 2 * lhalf) * I + n0 + lidx;
            const float* wup = Wu + (size_t)(kc + 2 * lhalf) * I + n0 + lidx;
            for (int k0 = 0; k0 < KC; k0 += 4) {
                v2f a0, a1, bg, bu;
                a0[0] = a0p[0]; a0[1] = a0p[1];
                a1[0] = a1p[0]; a1[1] = a1p[1];
                bg[0] = wgp[0]; bg[1] = wgp[I];
                bu[0] = wup[0]; bu[1] = wup[I];
                cg[0][nt] = __builtin_amdgcn_wmma_f32_16x16x4_f32(
                    false, a0, false, bg, (short)0, cg[0][nt], false, false);
                cg[1][nt] = __builtin_amdgcn_wmma_f32_16x16x4_f32(
                    false, a1, false, bg, (short)0, cg[1][nt], false, false);
                cu[0][nt] = __builtin_amdgcn_wmma_f32_16x16x4_f32(
                    false, a0, false, bu, (short)0, cu[0][nt], false, false);
                cu[1][nt] = __builtin_amdgcn_wmma_f32_16x16x4_f32(
                    false, a1, false, bu, (short)0, cu[1][nt], false, false);
                a0p += 4; a1p += 4;
                wgp += (size_t)4 * I;
                wup += (size_t)4 * I;
            }
        }
        __syncthreads();    // chunk done (also: last chunk -> xs dead, buf reusable)
    }

    // h = routing_weight * silu(g) * u, registers -> LDS h tile (aliases xs)
    #pragma unroll
    for (int ms = 0; ms < 2; ++ms) {
        #pragma unroll
        for (int nt = 0; nt < 4; ++nt) {
            const int n0 = c0 + nt * 16;
            #pragma unroll
            for (int j = 0; j < 8; ++j) {
                int   m = ms * 16 + j + 8 * lhalf;
                float g = cg[ms][nt][j];
                float u = cu[ms][nt][j];
                float s = g / (1.f + __expf(-g));
                buf[m * LSTR + n0 + lidx] = wrow[m] * s * u;
            }
        }
    }
    __syncthreads();

    // ---- Phase B: y = h(MTxI) @ w_down(IxH); wave owns 128 H-columns ----
    const float* Wd = w_down + (size_t)e * I * H;
    const int d0 = wave * 128;
    for (int nt = 0; nt < 8; ++nt) {
        const int n0 = d0 + nt * 16;
        v8f y0 = (v8f){}, y1 = (v8f){};
        const float* a0p = buf + lidx * LSTR + 2 * lhalf;
        const float* a1p = a0p + 16 * LSTR;
        const float* wp  = Wd + (size_t)(2 * lhalf) * H + n0 + lidx;
        for (int k0 = 0; k0 < I; k0 += 4) {
            v2f a0, a1, b;
            a0[0] = a0p[0]; a0[1] = a0p[1];
            a1[0] = a1p[0]; a1[1] = a1p[1];
            b[0]  = wp[0];  b[1]  = wp[H];
            y0 = __builtin_amdgcn_wmma_f32_16x16x4_f32(
                false, a0, false, b, (short)0, y0, false, false);
            y1 = __builtin_amdgcn_wmma_f32_16x16x4_f32(
                false, a1, false, b, (short)0, y1, false, false);
            a0p += 4; a1p += 4;
            wp  += (size_t)4 * H;
        }
        #pragma unroll
        for (int ms = 0; ms < 2; ++ms) {
            const v8f& c = ms ? y1 : y0;
            #pragma unroll
            for (int j = 0; j < 8; ++j) {
                int m = ms * 16 + j + 8 * lhalf;
                if (m < mrows) {   // guard stores only; WMMA ran under full EXEC
                    size_t o = ((size_t)trow[m] * TOPK + srow[m]) * H + n0 + lidx;
                    ybuf[o] = c[j];
                }
            }
        }
    }
}

// ---------------- Combine: out[t,h] = sum_k ybuf[t,k,h] (fixed order) ----------------
__global__ __launch_bounds__(256) void combine_kernel(
    const float* __restrict__ ybuf, float* __restrict__ out)
{
    int i = blockIdx.x * 256 + threadIdx.x;          // over T*H/4 float4s
    if (i >= T * H / 4) return;
    int t = i / (H / 4), c4 = (i % (H / 4)) * 4;
    const float4* p = (const float4*)(ybuf + (size_t)t * TOPK * H + c4);
    float4 a = p[0];
    float4 b = p[H / 4];
    float4 c = p[2 * (H / 4)];
    float4 d = p[3 * (H / 4)];
    float4 r;
    r.x = (a.x + b.x) + (c.x + d.x);
    r.y = (a.y + b.y) + (c.y + d.y);
    r.z = (a.z + b.z) + (c.z + d.z);
    r.w = (a.w + b.w) + (c.w + d.w);
    *(float4*)(out + (size_t)t * H + c4) = r;
}

extern "C" void kernel_launch(void* const* d_in, const int* in_sizes, int n_in,
                              void* d_out, int out_size, void* d_ws, size_t ws_size,
                              hipStream_t stream)
{
    const float* x  = (const float*)d_in[0];   // [T, H]
    const float* gw = (const float*)d_in[1];   // [E, H]
    const float* wg = (const float*)d_in[2];   // [E, H, I]
    const float* wu = (const float*)d_in[3];   // [E, H, I]
    const float* wd = (const float*)d_in[4];   // [E, I, H]
    const float* eb = (const float*)d_in[5];   // [E]
    // d_in[6] = top_k (hardcoded 4)

    char* ws = (char*)d_ws;
    size_t off = 0;
    int*   counts = (int*)(ws + off);   off += 256;                    // 16 ints, padded
    int*   sel    = (int*)(ws + off);   off += (size_t)T * TOPK * 4;   // 16 KB
    float* rw     = (float*)(ws + off); off += (size_t)T * TOPK * 4;   // 16 KB
    int*   ltok   = (int*)(ws + off);   off += (size_t)E * T * 4;      // 64 KB
    int*   lslot  = (int*)(ws + off);   off += (size_t)E * T * 4;      // 64 KB
    float* ybuf   = (float*)(ws + off);                                // 16 MB

    float* out = (float*)d_out;

    const size_t smem = (size_t)MT * LSTR * sizeof(float);   // ~66 KB dynamic LDS

    router_kernel <<<T, 32, 0, stream>>>(x, gw, eb, sel, rw);
    zero_kernel   <<<1, 32, 0, stream>>>(counts);
    scatter_kernel<<<T / 256, 256, 0, stream>>>(sel, counts, ltok, lslot);
    expert_kernel <<<E * 32, 256, smem, stream>>>(x, wg, wu, wd, rw,
                                                  counts, ltok, lslot, ybuf);
    combine_kernel<<<(T * H / 4 + 255) / 256, 256, 0, stream>>>(ybuf, out);
}